// RegionLoss_1Class_14439680049763
// MI455X (gfx1250) — compile-verified
//
#include <hip/hip_runtime.h>
#include <hip/hip_bf16.h>

#define W_    38
#define H_    38
#define A_    25
#define HW_   1444     // 38*38
#define HW4_  361      // HW_/4
#define NBLOCKS_MAIN 2048

typedef float v2f __attribute__((ext_vector_type(2)));
typedef float v8f __attribute__((ext_vector_type(8)));

// ANCHORS = _BASE * s, s in {0.5,0.75,1.0,1.25,1.5}; anchor a = (base[a%5], scale[a/5])
__constant__ float cBW[5] = {1.3221f, 3.19275f, 5.05587f, 9.47112f, 11.2364f};
__constant__ float cBH[5] = {1.73145f, 4.00944f, 8.09892f, 4.84053f, 10.0071f};
__constant__ float cSC[5] = {0.5f, 0.75f, 1.0f, 1.25f, 1.5f};

struct PerB {
    float gx, gy, gw, gh;     // gt box in grid units
    float tb0, tb1, tb2, tb3; // special-cell coord targets
    int   igi, igj, best, pad;
};

__device__ __forceinline__ float comp4(const float4& v, int e) {
    switch (e) { case 0: return v.x; case 1: return v.y; case 2: return v.z; default: return v.w; }
}

__device__ __forceinline__ float fast_rcp(float x) {
    return __builtin_amdgcn_rcpf(x);          // v_rcp_f32 (1 trans op, no IEEE div expansion)
}
__device__ __forceinline__ float fast_sigmoid(float x) {
    return fast_rcp(1.0f + __expf(-x));       // v_exp_f32 + v_rcp_f32
}

// ---------------- per-batch precompute ----------------
__global__ void __launch_bounds__(256)
prep_kernel(const float* __restrict__ target, PerB* __restrict__ pb, int B)
{
    int b = blockIdx.x * blockDim.x + threadIdx.x;
    if (b >= B) return;
    float gx = target[b * 4 + 0] * (float)W_;
    float gy = target[b * 4 + 1] * (float)H_;
    float gw = target[b * 4 + 2] * (float)W_;
    float gh = target[b * 4 + 3] * (float)H_;
    int gi = (int)gx, gj = (int)gy;

    // anchor-vs-gt IoU (both centered at origin): inter = min(aw,gw)*min(ah,gh)
    int best = 0; float best_iou = -1.0f;
    #pragma unroll
    for (int a = 0; a < A_; ++a) {
        float aw = cBW[a % 5] * cSC[a / 5];
        float ah = cBH[a % 5] * cSC[a / 5];
        float inter = fminf(aw, gw) * fminf(ah, gh);
        float uni   = aw * ah + gw * gh - inter;
        float iou   = inter * fast_rcp(uni);
        if (iou > best_iou) { best_iou = iou; best = a; }  // first max wins
    }
    float baw = cBW[best % 5] * cSC[best / 5];
    float bah = cBH[best % 5] * cSC[best / 5];

    PerB r;
    r.gx = gx; r.gy = gy; r.gw = gw; r.gh = gh;
    r.tb0 = gx - (float)gi;
    r.tb1 = gy - (float)gj;
    r.tb2 = __logf(gw * fast_rcp(baw));
    r.tb3 = __logf(gh * fast_rcp(bah));
    r.igi = gi; r.igj = gj; r.best = best; r.pad = 0;
    pb[b] = r;
}

// ---------------- main streaming loss kernel ----------------
__global__ void __launch_bounds__(256)
region_loss_main(const float* __restrict__ pred, const PerB* __restrict__ pb,
                 float* __restrict__ partials, int total4)
{
    const int tid    = blockIdx.x * blockDim.x + threadIdx.x;
    const int stride = gridDim.x * blockDim.x;

    float lsum = 0.0f;

    for (int idx = tid; idx < total4; idx += stride) {
        int b   = idx / (A_ * HW4_);
        int rem = idx - b * (A_ * HW4_);
        int a   = rem / HW4_;
        int hw4 = rem - a * HW4_;

        const float* base = pred + ((size_t)(b * (A_ * 5) + a * 5)) * HW_ + (size_t)hw4 * 4;

        float4 q[5];
        #pragma unroll
        for (int k = 0; k < 5; ++k)
            q[k] = *reinterpret_cast<const float4*>(base + k * HW_);   // global_load_b128 x5

        PerB pv = pb[b];
        float aw = cBW[a % 5] * cSC[a / 5];
        float ah = cBH[a % 5] * cSC[a / 5];
        float gx0 = pv.gx - pv.gw * 0.5f, gx1 = pv.gx + pv.gw * 0.5f;
        float gy0 = pv.gy - pv.gh * 0.5f, gy1 = pv.gy + pv.gh * 0.5f;
        float garea = pv.gw * pv.gh;
        bool  abest = (a == pv.best);

        #pragma unroll
        for (int e = 0; e < 4; ++e) {
            int hw = hw4 * 4 + e;
            int i  = hw % W_;
            int j  = hw / W_;
            float p0 = comp4(q[0], e), p1 = comp4(q[1], e), p2 = comp4(q[2], e);
            float p3 = comp4(q[3], e), p4 = comp4(q[4], e);

            float tx = fast_sigmoid(p0);
            float ty = fast_sigmoid(p1);
            float pc = fast_sigmoid(p4);
            float bw = __expf(p2) * aw;
            float bh = __expf(p3) * ah;
            float bx = tx + (float)i;
            float by = ty + (float)j;

            float mx = fminf(bx - bw * 0.5f, gx0);
            float Mx = fmaxf(bx + bw * 0.5f, gx1);
            float my = fminf(by - bh * 0.5f, gy0);
            float My = fmaxf(by + bh * 0.5f, gy1);
            float cw = bw + pv.gw - (Mx - mx);
            float ch = bh + pv.gh - (My - my);
            float inter = (cw > 0.0f && ch > 0.0f) ? cw * ch : 0.0f;
            float uni   = bw * bh + garea - inter;
            float iou   = inter * fast_rcp(uni);

            bool special = abest && (i == pv.igi) && (j == pv.igj);
            float wc = special ? 5.0f : (iou > 0.6f ? 0.0f : 1.0f);  // conf_mask^2
            float t0 = special ? pv.tb0 : 0.5f;
            float t1 = special ? pv.tb1 : 0.5f;
            float t2 = special ? pv.tb2 : 0.0f;
            float t3 = special ? pv.tb3 : 0.0f;
            float tc = special ? iou    : 0.0f;

            float d0 = tx - t0, d1 = ty - t1, d2 = p2 - t2, d3 = p3 - t3, dc = pc - tc;
            lsum += 0.5f * (d0 * d0 + d1 * d1 + d2 * d2 + d3 * d3 + wc * dc * dc);
        }
    }

    // ---- wave32 reduction via V_WMMA_F32_16X16X4_F32 ----
    // A = ones(16x4), B holds the 32 lane partials (+32 zeros) in its 64 slots.
    // D[m][n] = sum_k B[k][n]  => summing 16 columns of any D row = total of all
    // B entries, independent of B's VGPR striping (layout-agnostic exact trick).
    v2f am; am.x = 1.0f; am.y = 1.0f;
    v2f bm; bm.x = lsum; bm.y = 0.0f;
    v8f cm = {};
    v8f dm = __builtin_amdgcn_wmma_f32_16x16x4_f32(
        /*neg_a=*/false, am, /*neg_b=*/false, bm,
        /*c_mod=*/(short)0, cm, /*reuse_a=*/false, /*reuse_b=*/false);

    float wsum = dm[0];                       // row-of-D component in this lane
    #pragma unroll
    for (int off = 1; off < 16; off <<= 1)    // sum the 16 column-sums
        wsum += __shfl_xor(wsum, off, 32);
    // lane 0 now holds the full wave sum

    __shared__ float sbuf[8];                 // 256 threads = 8 waves
    int lane = threadIdx.x & 31;
    int wv   = threadIdx.x >> 5;
    if (lane == 0) sbuf[wv] = wsum;
    __syncthreads();
    if (threadIdx.x == 0) {
        float s = 0.0f;
        #pragma unroll
        for (int w2 = 0; w2 < 8; ++w2) s += sbuf[w2];   // fixed order: deterministic
        partials[blockIdx.x] = s;
    }
}

// ---------------- deterministic final reduction ----------------
__global__ void __launch_bounds__(256)
final_reduce(const float* __restrict__ partials, int n, float* __restrict__ out)
{
    __shared__ float sbuf[8];
    float s = 0.0f;
    for (int i = threadIdx.x; i < n; i += blockDim.x) s += partials[i];
    #pragma unroll
    for (int off = 16; off; off >>= 1) s += __shfl_xor(s, off, 32);
    int lane = threadIdx.x & 31;
    int wv   = threadIdx.x >> 5;
    if (lane == 0) sbuf[wv] = s;
    __syncthreads();
    if (threadIdx.x == 0) {
        float t = 0.0f;
        #pragma unroll
        for (int w2 = 0; w2 < 8; ++w2) t += sbuf[w2];
        out[0] = t;
    }
}

extern "C" void kernel_launch(void* const* d_in, const int* in_sizes, int n_in,
                              void* d_out, int out_size, void* d_ws, size_t ws_size,
                              hipStream_t stream) {
    const float* pred   = (const float*)d_in[0];
    const float* target = (const float*)d_in[1];
    (void)n_in; (void)out_size; (void)ws_size; // train_out (d_in[2]) == 0: scalar loss only

    const int B = in_sizes[1] / 4;            // 512
    PerB*  pb       = (PerB*)d_ws;
    float* partials = (float*)((char*)d_ws + (size_t)B * sizeof(PerB));

    prep_kernel<<<(B + 255) / 256, 256, 0, stream>>>(target, pb, B);

    const int total4 = B * A_ * HW4_;         // 4,620,800 work items (x4 elems each)
    region_loss_main<<<NBLOCKS_MAIN, 256, 0, stream>>>(pred, pb, partials, total4);
    final_reduce<<<1, 256, 0, stream>>>(partials, NBLOCKS_MAIN, (float*)d_out);
}